// FPNPredictor_28535762715386
// MI455X (gfx1250) — compile-verified
//
#include <hip/hip_runtime.h>

typedef __attribute__((ext_vector_type(2))) float v2f;
typedef __attribute__((ext_vector_type(8))) float v8f;

#define NROWS 65536
#define DDIM  1024
#define FDIM  300
#define NCLS  81      // scores cols (1 + 80)
#define NBB   324     // bbox cols
#define NPAD  640     // 1 + 300 + 324 = 625 padded to 640

// workspace layout in floats
#define OFF_WALL 0
#define OFF_BIAS (NPAD * DDIM)              // 655360
#define OFF_SM   (OFF_BIAS + NPAD)          // 656000
#define OFF_SEM  (OFF_SM + 80 * FDIM)       // 680000

static __device__ __forceinline__ v2f mk2(float a, float b) {
    v2f r; r.x = a; r.y = b; return r;
}

// Async DMA: copy 8 bytes global -> LDS, tracked by ASYNCcnt.
// lds_off: raw 32-bit LDS byte address (low 32 bits of flat shared pointer).
static __device__ __forceinline__ void async_b64(unsigned lds_off, const float* g) {
    asm volatile("global_load_async_to_lds_b64 %0, %1, off"
                 :: "v"(lds_off), "v"(g)
                 : "memory");
}

// ---------------------------------------------------------------------------
// Kernel 1: pack [W_cls; W_sem; W_bbox] -> W_all[640][1024], biases -> ball.
// ---------------------------------------------------------------------------
__global__ __launch_bounds__(256) void pack_weights(
    const float* __restrict__ Wc, const float* __restrict__ bc,
    const float* __restrict__ Ws, const float* __restrict__ bs,
    const float* __restrict__ Wb, const float* __restrict__ bb,
    float* __restrict__ wall, float* __restrict__ ball) {
    const int n = blockIdx.x;
    const int t = threadIdx.x;
    const float* src = nullptr;
    float bias = 0.0f;
    if (n == 0)        { src = Wc;                    bias = bc[0]; }
    else if (n <= 300) { src = Ws + (n - 1) * DDIM;   bias = bs[n - 1]; }
    else if (n <= 624) { src = Wb + (n - 301) * DDIM; bias = bb[n - 301]; }
    float4 v = make_float4(0.f, 0.f, 0.f, 0.f);
    if (src) v = reinterpret_cast<const float4*>(src)[t];
    reinterpret_cast<float4*>(wall + (long)n * DDIM)[t] = v;
    if (t == 0) ball[n] = bias;
}

// ---------------------------------------------------------------------------
// Kernel 2: sm = 8 * sem_matrix / ||row||_2.  One block per row (80 rows).
// ---------------------------------------------------------------------------
__global__ __launch_bounds__(256) void make_sm(
    const float* __restrict__ smat, float* __restrict__ sm) {
    __shared__ float red[256];
    __shared__ float scale_s;
    const int r = blockIdx.x;
    const int t = threadIdx.x;
    float acc = 0.f;
    for (int j = t; j < FDIM; j += 256) {
        float v = smat[(long)r * FDIM + j];
        acc += v * v;
    }
    red[t] = acc;
    __syncthreads();
    for (int s = 128; s > 0; s >>= 1) {
        if (t < s) red[t] += red[t + s];
        __syncthreads();
    }
    if (t == 0) scale_s = 8.0f * rsqrtf(red[0]);
    __syncthreads();
    const float sc = scale_s;
    for (int j = t; j < FDIM; j += 256)
        sm[(long)r * FDIM + j] = smat[(long)r * FDIM + j] * sc;
}

// ---------------------------------------------------------------------------
// Kernel 3: fused GEMM  out[640 cols] = x[N,1024] @ W_all.T + bias
//   WG: 256 threads = 8 waves. Tile M=64, N=160, KC=32, double-buffered LDS.
//   Global->LDS staging uses GLOBAL_LOAD_ASYNC_TO_LDS_B64 (ASYNCcnt) with a
//   per-lane pair-swizzled layout: entry [p][r] = (k=2p, k=2p+1) so a lane's
//   WMMA fragment (VGPR0:K=2*half, VGPR1:K=2*half+1) is one b64 DS read.
// ---------------------------------------------------------------------------
__global__ __launch_bounds__(256) void fused_gemm(
    const float* __restrict__ x, const float* __restrict__ wall,
    const float* __restrict__ ball, float* __restrict__ out,
    float* __restrict__ sem) {
    __shared__ v2f Alds[2 * 16 * 64];    // double-buffered [pair][row]
    __shared__ v2f Blds[2 * 16 * 160];   // double-buffered [pair][col]

    const int tid   = threadIdx.x;
    const int wave  = tid >> 5;
    const int lid   = tid & 31;
    const int half  = lid >> 4;
    const int ln    = lid & 15;
    const int mTile = wave & 3;
    const int nHalf = wave >> 2;
    const int rowBase = blockIdx.y * 64;
    const int colBase = blockIdx.x * 160;

    // ---- per-thread async-copy address setup (computed once) ----
    const unsigned aBase = (unsigned)(uintptr_t)(&Alds[0]);
    const unsigned bBase = (unsigned)(uintptr_t)(&Blds[0]);
    const int rA = tid & 63, pA = tid >> 6;       // A slot: pair pA+4i, row rA
    unsigned aOff[4];  const float* aG[4];
    unsigned bOff[10]; const float* bG[10];
#pragma unroll
    for (int i = 0; i < 4; ++i) {
        const int p = pA + 4 * i;
        aOff[i] = aBase + (unsigned)((p * 64 + rA) * 8);
        aG[i]   = x + (long)(rowBase + rA) * DDIM + 2 * p;
    }
#pragma unroll
    for (int i = 0; i < 10; ++i) {
        const int s = tid + i * 256;              // 0..2559
        const int p = s / 160;
        const int n = s - p * 160;
        bOff[i] = bBase + (unsigned)((p * 160 + n) * 8);
        bG[i]   = wall + (long)(colBase + n) * DDIM + 2 * p;
    }

    const v8f zero = {0.f, 0.f, 0.f, 0.f, 0.f, 0.f, 0.f, 0.f};
    v8f acc[5];
#pragma unroll
    for (int i = 0; i < 5; ++i) acc[i] = zero;

    // issue chunk 0 into buffer 0
#pragma unroll
    for (int i = 0; i < 4; ++i)  async_b64(aOff[i], aG[i]);
#pragma unroll
    for (int i = 0; i < 10; ++i) async_b64(bOff[i], bG[i]);

    for (int it = 0; it < 32; ++it) {
        const int buf = it & 1;
        if (it < 31) {
            // issue chunk it+1 into the other buffer (14 async b64 / thread)
            const int k0n = (it + 1) * 32;
            const unsigned ao = (buf ^ 1) ? 8192u : 0u;      // 1024 v2f * 8B
            const unsigned bo = (buf ^ 1) ? 20480u : 0u;     // 2560 v2f * 8B
#pragma unroll
            for (int i = 0; i < 4; ++i)  async_b64(aOff[i] + ao, aG[i] + k0n);
#pragma unroll
            for (int i = 0; i < 10; ++i) async_b64(bOff[i] + bo, bG[i] + k0n);
            // previous 14 (chunk it) are done once <=14 remain outstanding
            asm volatile("s_wait_asynccnt 0xe" ::: "memory");
        } else {
            asm volatile("s_wait_asynccnt 0x0" ::: "memory");
        }
        __syncthreads();

        const v2f* Ab = Alds + buf * 1024;
        const v2f* Bb = Blds + buf * 2560;

        // software-pipelined 8 k-steps x 5 column tiles
        v2f a_cur = Ab[half * 64 + mTile * 16 + ln];
        v2f b_cur[5];
#pragma unroll
        for (int tn = 0; tn < 5; ++tn)
            b_cur[tn] = Bb[half * 160 + nHalf * 80 + tn * 16 + ln];

#pragma unroll
        for (int ks = 0; ks < 8; ++ks) {
            v2f a_nxt = a_cur;
            v2f b_nxt[5];
#pragma unroll
            for (int tn = 0; tn < 5; ++tn) b_nxt[tn] = b_cur[tn];
            if (ks < 7) {
                const int pr = (ks + 1) * 2 + half;
                a_nxt = Ab[pr * 64 + mTile * 16 + ln];
#pragma unroll
                for (int tn = 0; tn < 5; ++tn)
                    b_nxt[tn] = Bb[pr * 160 + nHalf * 80 + tn * 16 + ln];
            }
#pragma unroll
            for (int tn = 0; tn < 5; ++tn) {
                acc[tn] = __builtin_amdgcn_wmma_f32_16x16x4_f32(
                    false, a_cur, false, b_cur[tn], (short)0, acc[tn], false, false);
            }
            a_cur = a_nxt;
#pragma unroll
            for (int tn = 0; tn < 5; ++tn) b_cur[tn] = b_nxt[tn];
        }
        __syncthreads();
    }

    // epilogue: C/D layout -> lane ln = column, VGPR i = row (half*8 + i)
    const int row0 = rowBase + mTile * 16 + half * 8;
#pragma unroll
    for (int tn = 0; tn < 5; ++tn) {
        const int c = colBase + nHalf * 80 + tn * 16 + ln;
        const float bias = ball[c];
        float* dst = nullptr;
        int stride = 0;
        if (c == 0)         { dst = out + (long)row0 * NCLS;                         stride = NCLS; }
        else if (c <= 300)  { dst = sem + (long)row0 * FDIM + (c - 1);               stride = FDIM; }
        else if (c <= 624)  { dst = out + (long)NROWS * NCLS + (long)row0 * NBB + (c - 301); stride = NBB; }
        if (dst) {
#pragma unroll
            for (int i = 0; i < 8; ++i) dst[(long)i * stride] = acc[tn][i] + bias;
        }
    }
}

// ---------------------------------------------------------------------------
// Kernel 4: cls_score = sem[N,300] @ sm.T[300,80] -> scores[:,1:81]
// ---------------------------------------------------------------------------
__global__ __launch_bounds__(128) void cls_gemm(
    const float* __restrict__ sem, const float* __restrict__ sm,
    float* __restrict__ out) {
    const int tid  = threadIdx.x;
    const int wave = tid >> 5;
    const int lid  = tid & 31;
    const int half = lid >> 4;
    const int ln   = lid & 15;
    const int rowBase = blockIdx.x * 64 + wave * 16;

    const v8f zero = {0.f, 0.f, 0.f, 0.f, 0.f, 0.f, 0.f, 0.f};
    v8f acc[5];
#pragma unroll
    for (int i = 0; i < 5; ++i) acc[i] = zero;

    const float* arow = sem + (long)(rowBase + ln) * FDIM;

    for (int ks = 0; ks < 75; ++ks) {
        const int k = ks * 4 + half * 2;
        const v2f a = *reinterpret_cast<const v2f*>(arow + k);
#pragma unroll
        for (int tn = 0; tn < 5; ++tn) {
            const v2f b = *reinterpret_cast<const v2f*>(
                sm + (long)(tn * 16 + ln) * FDIM + k);
            acc[tn] = __builtin_amdgcn_wmma_f32_16x16x4_f32(
                false, a, false, b, (short)0, acc[tn], false, false);
        }
    }

    const int row0 = rowBase + half * 8;
#pragma unroll
    for (int tn = 0; tn < 5; ++tn) {
        const int c = tn * 16 + ln;           // 0..79 -> scores col 1+c
        float* dst = out + (long)row0 * NCLS + 1 + c;
#pragma unroll
        for (int i = 0; i < 8; ++i) dst[(long)i * NCLS] = acc[tn][i];
    }
}

// ---------------------------------------------------------------------------
extern "C" void kernel_launch(void* const* d_in, const int* in_sizes, int n_in,
                              void* d_out, int out_size, void* d_ws, size_t ws_size,
                              hipStream_t stream) {
    (void)in_sizes; (void)n_in; (void)out_size; (void)ws_size;
    const float* x    = (const float*)d_in[0];
    const float* Wc   = (const float*)d_in[1];
    const float* bc   = (const float*)d_in[2];
    const float* Ws   = (const float*)d_in[3];
    const float* bs   = (const float*)d_in[4];
    const float* Wb   = (const float*)d_in[5];
    const float* bb   = (const float*)d_in[6];
    const float* smat = (const float*)d_in[7];
    float* out = (float*)d_out;
    float* ws  = (float*)d_ws;

    float* wall = ws + OFF_WALL;
    float* ball = ws + OFF_BIAS;
    float* sm   = ws + OFF_SM;
    float* sem  = ws + OFF_SEM;

    pack_weights<<<dim3(NPAD), 256, 0, stream>>>(Wc, bc, Ws, bs, Wb, bb, wall, ball);
    make_sm<<<dim3(80), 256, 0, stream>>>(smat, sm);
    fused_gemm<<<dim3(NPAD / 160, NROWS / 64), 256, 0, stream>>>(x, wall, ball, out, sem);
    cls_gemm<<<dim3(NROWS / 64), 128, 0, stream>>>(sem, sm, out);
}